// TopKActivation_36764920054293
// MI455X (gfx1250) — compile-verified
//
#include <hip/hip_runtime.h>
#include <stdint.h>

// TopK activation: out = relu(x) masked to values >= (k-th largest relu(x) per row).
// Rows = 16384, Cols = 8192 (fp32), k = 256 (read from device scalar).
//
// 1 workgroup (256 thr = 8 waves) per row.
//  A) Row staged global->LDS with CDNA5 async copy (global_load_async_to_lds_b128,
//     ASYNCcnt, s_wait_asynccnt) -- single global read.
//  B) ReLU bit patterns hoisted to a 32-entry register array (one conflict-free
//     LDS sweep). Exact k-th largest via 4-pass MSB-first radix select on uint
//     patterns (valid: relu >= 0 orders like uint). LDS ds_add_u32 histograms;
//     zeros skipped (threshold==0 case detected explicitly). Bin scan is
//     parallel: wave0, 8 bins/lane, __shfl_down suffix scan across lanes.
//  C) Masked output from registers via non-temporal stores (write-once stream).

#define COLS 8192
#define BLOCK 256
#define PER_THREAD (COLS / BLOCK)   // 32 elements per thread

__global__ __launch_bounds__(BLOCK)
void TopKActivation_36764920054293_kernel(const float* __restrict__ x,
                                          const int* __restrict__ kptr,
                                          float* __restrict__ out,
                                          int rows) {
    __shared__ __align__(16) float row[COLS];   // 32 KB async-copy landing buffer
    __shared__ unsigned int hist[256];
    __shared__ unsigned int sh_prefix;
    __shared__ unsigned int sh_kremain;
    __shared__ unsigned int sh_done;

    const int tid = threadIdx.x;
    const int r   = blockIdx.x;
    if (r >= rows) return;                       // uniform per block

    const float* __restrict__ src = x   + (size_t)r * COLS;
    float*       __restrict__ dst = out + (size_t)r * COLS;

    // ---- Phase A: async bulk copy global row -> LDS (16B/lane/issue) ----
    {
        // Low 32 bits of flat LDS address == workgroup-relative DS byte offset.
        const uint32_t lds_base = (uint32_t)(uintptr_t)(&row[0]);
#pragma unroll
        for (int i = 0; i < PER_THREAD / 4; ++i) {          // 8 issues x 16B
            const int byte_off = (i * BLOCK + tid) * 16;
            const uint32_t lds_off = lds_base + (uint32_t)byte_off;
            const char* gp = (const char*)src + byte_off;
            asm volatile("global_load_async_to_lds_b128 %0, %1, off"
                         :
                         : "v"(lds_off), "v"(gp)
                         : "memory");
        }
        asm volatile("s_wait_asynccnt 0" ::: "memory");
    }

    int k = *kptr;                    // uniform scalar load
    if (k > COLS) k = COLS;
    if (k < 1)    k = 1;

    if (tid == 0) {
        sh_prefix  = 0u;
        sh_kremain = (unsigned int)k;
        sh_done    = 0u;
    }
    __syncthreads();                  // LDS row visible, scalars initialized

    // ---- Single LDS sweep -> register-resident relu bit patterns ----
    uint32_t v[PER_THREAD];
#pragma unroll
    for (int j = 0; j < PER_THREAD; ++j) {
        const float f = row[j * BLOCK + tid];   // lane-stride-1: no bank conflicts
        v[j] = (f > 0.0f) ? __float_as_uint(f) : 0u;
    }

    // ---- Phase B: 4-pass radix select (MSB-first, 8 bits/pass) ----
    for (int pass = 0; pass < 4; ++pass) {
        const int shift = 24 - pass * 8;
        hist[tid] = 0u;
        __syncthreads();

        if (sh_done) break;                     // uniform: threshold is exactly 0

        const uint32_t prefix = sh_prefix;
        const uint32_t himask = (pass == 0) ? 0u
                                            : (0xFFFFFFFFu << (32 - pass * 8));
#pragma unroll
        for (int j = 0; j < PER_THREAD; ++j) {
            const uint32_t b = v[j];
            if (b != 0u && (b & himask) == prefix) {   // zeros never histogrammed
                atomicAdd(&hist[(b >> shift) & 0xFFu], 1u);
            }
        }
        __syncthreads();

        // Parallel bin scan: wave 0, 8 bins per lane, suffix sums via shuffles.
        if (tid < 32) {
            const int lane = tid;
            unsigned c[8];
            unsigned lane_total = 0;
#pragma unroll
            for (int j = 0; j < 8; ++j) {
                c[j] = hist[lane * 8 + j];
                lane_total += c[j];
            }
            // Inclusive suffix sum over lanes (higher lane = higher bins).
            unsigned s = lane_total;
#pragma unroll
            for (int off = 1; off < 32; off <<= 1) {
                const unsigned u = __shfl_down(s, off, 32);
                if (lane + off < 32) s += u;
            }
            const unsigned above = s - lane_total;     // strictly-higher lanes
            const unsigned krem  = sh_kremain;

            if (lane == 0 && s < krem) {
                // Fewer qualifying (positive) elements than rank -> k-th value is 0.
                sh_done   = 1u;
                sh_prefix = 0u;
            }
            if (above < krem && krem <= above + lane_total) {
                // Exactly one lane lands here: locate digit within its 8 bins.
                unsigned cum = above, d = 0, nk = 0;
#pragma unroll
                for (int j = 7; j >= 0; --j) {
                    if (cum + c[j] >= krem) {
                        d  = (unsigned)(lane * 8 + j);
                        nk = krem - cum;
                        break;
                    }
                    cum += c[j];
                }
                sh_prefix  = prefix | (d << shift);
                sh_kremain = nk;
            }
        }
        __syncthreads();
    }

    // Full 32-bit pattern of the k-th largest relu value (0 if rank falls in zeros).
    const uint32_t t = sh_done ? 0u : sh_prefix;

    // ---- Phase C: masked output from registers, non-temporal streaming stores ----
#pragma unroll
    for (int j = 0; j < PER_THREAD; ++j) {
        const int i = j * BLOCK + tid;          // lane-contiguous: coalesced
        const uint32_t b = v[j];
        const float o = (b >= t) ? __uint_as_float(b) : 0.0f;
        __builtin_nontemporal_store(o, &dst[i]);
    }
}

extern "C" void kernel_launch(void* const* d_in, const int* in_sizes, int n_in,
                              void* d_out, int out_size, void* d_ws, size_t ws_size,
                              hipStream_t stream) {
    const float* x  = (const float*)d_in[0];
    const int*   kp = (const int*)d_in[1];
    float*       o  = (float*)d_out;

    const int rows = in_sizes[0] / COLS;

    dim3 grid(rows), block(BLOCK);
    hipLaunchKernelGGL(TopKActivation_36764920054293_kernel,
                       grid, block, 0, stream, x, kp, o, rows);
}